// SSDChunkwiseScan_17068200034421
// MI455X (gfx1250) — compile-verified
//
#include <hip/hip_runtime.h>
#include <stdint.h>

// Problem constants (B, L, H, D from reference setup_inputs; chunk = 64)
#define B_    4
#define L_    4096
#define H_    16
#define D_    64
#define C_    64
#define NCH_  (L_ / C_)

typedef float v2f __attribute__((ext_vector_type(2)));
typedef float v8f __attribute__((ext_vector_type(8)));

// A_bar as complex w = p + i q acting on state (x,y): x' = p x + q y ; y' = -q x + p y
// (matches reference row0=[a11,a12], row1=[-a12,a11]; composition == complex multiply)
__device__ __forceinline__ void make_w(float a, float o, float dv, float& p, float& q) {
  float tau = 0.5f * dv;
  float ta  = tau * a;
  float to  = tau * o;
  float den = (1.0f + ta) * (1.0f + ta) + to * to + 1e-6f;
  float inv = 1.0f / den;
  p = ((1.0f + ta) * (1.0f - ta) - to * to) * inv;   // a11
  q = 2.0f * to * inv;                               // a12
}

#define USE_ASYNC_LDS 1
#if USE_ASYNC_LDS
// CDNA5 async global->LDS DMA (per-lane VDST = LDS byte offset, VADDR = 64-bit global addr)
__device__ __forceinline__ void async_ld_f32(float* lds_dst, const float* gsrc) {
  uint32_t loff = (uint32_t)(uintptr_t)lds_dst;  // low 32 bits of flat LDS addr = DS offset
  asm volatile("global_load_async_to_lds_b32 %0, %1, off"
               :: "v"(loff), "v"(gsrc) : "memory");
}
__device__ __forceinline__ void wait_async0() {
  asm volatile("s_wait_asynccnt 0" ::: "memory");
}
#endif

// Complex WMMA step: accX += aP@bx + aQ@by ; accY += aP@by - aQ@bx
__device__ __forceinline__ void cwmma(const v2f& aP, const v2f& aQ, const v2f& aQn,
                                      const v2f& bx, const v2f& by,
                                      v8f& accX, v8f& accY) {
  accX = __builtin_amdgcn_wmma_f32_16x16x4_f32(false, aP,  false, bx, (short)0, accX, false, false);
  accX = __builtin_amdgcn_wmma_f32_16x16x4_f32(false, aQ,  false, by, (short)0, accX, false, false);
  accY = __builtin_amdgcn_wmma_f32_16x16x4_f32(false, aP,  false, by, (short)0, accY, false, false);
  accY = __builtin_amdgcn_wmma_f32_16x16x4_f32(false, aQn, false, bx, (short)0, accY, false, false);
}

// ---------------- Phase A: per-chunk totals ----------------
// grid = B*NCH*H, block = 64 (thread = d channel)
__global__ void __launch_bounds__(C_)
ssd_phaseA(const float* __restrict__ alpha, const float* __restrict__ omega,
           const float* __restrict__ dt,    const float* __restrict__ K,
           const float* __restrict__ V,
           float* __restrict__ totw, float* __restrict__ finh) {
  int blk = blockIdx.x;
  int h  = blk % H_;
  int bn = blk / H_;
  int n  = bn % NCH_;
  int b  = bn / NCH_;
  int t0 = n * C_;
  int tid = threadIdx.x;

  __shared__ float wpS[C_], wqS[C_], dtS[C_];
  {
    int idx = (b * L_ + t0 + tid) * H_ + h;
    float p, q;
    float dv = dt[idx];
    make_w(alpha[idx], omega[idx], dv, p, q);
    wpS[tid] = p; wqS[tid] = q; dtS[tid] = dv;
  }
  __syncthreads();

  // sequential in-chunk scan for channel d = tid (exact recurrence)
  int d = tid;
  float sx = 0.0f, sy = 0.0f;
  for (int t = 0; t < C_; ++t) {
    int base = ((b * L_ + t0 + t) * H_ + h) * D_ + d;
    v2f kv = *(const v2f*)(K + 2 * base);
    float u  = V[base] * dtS[t];
    float p = wpS[t], q = wqS[t];
    float nx =  p * sx + q * sy + kv.x * u;
    float ny = -q * sx + p * sy + kv.y * u;
    sx = nx; sy = ny;
  }
  int fidx = (((b * NCH_ + n) * H_ + h) * D_ + d) * 2;
  finh[fidx]     = sx;
  finh[fidx + 1] = sy;

  if (tid == 0) {  // chunk total transition (complex product; commutative)
    float P = 1.0f, Q = 0.0f;
    for (int t = 0; t < C_; ++t) {
      float p = wpS[t], q = wqS[t];
      float nP = P * p - Q * q;
      float nQ = P * q + Q * p;
      P = nP; Q = nQ;
    }
    int widx = ((b * NCH_ + n) * H_ + h) * 2;
    totw[widx] = P; totw[widx + 1] = Q;
  }
}

// ---------------- Phase B: scan across chunks ----------------
// grid = B*H, block = 64 (thread = d); cstate[n] = state BEFORE chunk n
__global__ void __launch_bounds__(D_)
ssd_phaseB(const float* __restrict__ totw, const float* __restrict__ finh,
           float* __restrict__ cstate) {
  int h = blockIdx.x % H_;
  int b = blockIdx.x / H_;
  int d = threadIdx.x;
  float sx = 0.0f, sy = 0.0f;
  for (int n = 0; n < NCH_; ++n) {
    int base = (((b * NCH_ + n) * H_ + h) * D_ + d) * 2;
    cstate[base]     = sx;
    cstate[base + 1] = sy;
    int widx = ((b * NCH_ + n) * H_ + h) * 2;
    float P = totw[widx], Q = totw[widx + 1];
    float fx = finh[base], fy = finh[base + 1];
    float nx =  P * sx + Q * sy + fx;
    float ny = -Q * sx + P * sy + fy;
    sx = nx; sy = ny;
  }
}

// ---------------- Phase C: chunk matmul local_h = P @ U (WMMA) + correction ----------------
// grid = B*NCH*H, block = 128 (4 waves; wave w owns t-tile rows [16w,16w+16))
__global__ void __launch_bounds__(128)
ssd_phaseC(const float* __restrict__ alpha, const float* __restrict__ omega,
           const float* __restrict__ dt,    const float* __restrict__ K,
           const float* __restrict__ V,     const float* __restrict__ cstate,
           float* __restrict__ Y) {
  int blk = blockIdx.x;
  int h  = blk % H_;
  int bn = blk / H_;
  int n  = bn % NCH_;
  int b  = bn / NCH_;
  int t0g = n * C_;
  int tid = threadIdx.x;

  __shared__ float Ux[C_][D_ + 1];   // U real part, [s][d]
  __shared__ float Uy[C_][D_ + 1];   // U imag part
  __shared__ float aS[C_], oS[C_], dtS[C_];
  __shared__ float Wp[C_], Wq[C_];     // prefix product W_t (cum_A)
  __shared__ float IvP[C_], IvQ[C_];   // W_t^{-1} with reference's det+1e-8
  __shared__ float Sx[D_], Sy[D_];     // incoming chunk state

  // Stage alpha/omega/dt (strided gather) via CDNA5 async DMA; other waves load state.
  if (tid < C_) {
    int idx = (b * L_ + t0g + tid) * H_ + h;
#if USE_ASYNC_LDS
    async_ld_f32(&aS[tid],  alpha + idx);
    async_ld_f32(&oS[tid],  omega + idx);
    async_ld_f32(&dtS[tid], dt + idx);
#else
    aS[tid] = alpha[idx]; oS[tid] = omega[idx]; dtS[tid] = dt[idx];
#endif
  } else {
    int d = tid - C_;  // 0..63
    int sbase = (((b * NCH_ + n) * H_ + h) * D_ + d) * 2;
    Sx[d] = cstate[sbase];
    Sy[d] = cstate[sbase + 1];
  }
#if USE_ASYNC_LDS
  wait_async0();
#endif
  __syncthreads();

  if (tid < C_) {  // per-step w -> Wp/Wq (raw for now)
    float p, q;
    make_w(aS[tid], oS[tid], dtS[tid], p, q);
    Wp[tid] = p; Wq[tid] = q;
  }
  __syncthreads();
  if (tid == 0) {  // in-place prefix product (complex, 64 steps)
    float P = Wp[0], Q = Wq[0];
    for (int t = 1; t < C_; ++t) {
      float p = Wp[t], q = Wq[t];
      float nP = p * P - q * Q;
      float nQ = p * Q + q * P;
      P = nP; Q = nQ;
      Wp[t] = P; Wq[t] = Q;
    }
  }
  __syncthreads();
  if (tid < C_) {  // inverse as in reference: conj/(|W|^2 + 1e-8)
    float P = Wp[tid], Q = Wq[tid];
    float inv = 1.0f / (P * P + Q * Q + 1e-8f);
    IvP[tid] =  P * inv;
    IvQ[tid] = -Q * inv;
  }
  // Build U (64x64 complex) in LDS
  for (int i = tid; i < C_ * D_; i += 128) {
    int s = i >> 6, d = i & 63;
    int base = ((b * L_ + t0g + s) * H_ + h) * D_ + d;
    v2f kv = *(const v2f*)(K + 2 * base);
    float u = V[base] * dtS[s];
    Ux[s][d] = kv.x * u;
    Uy[s][d] = kv.y * u;
  }
  __syncthreads();

  // ---- WMMA: local_h = P @ U, complex via 4 real f32 16x16x4 matmuls ----
  int wave = tid >> 5;
  int lane = tid & 31;
  int m  = lane & 15;     // A row / B,C,D column within tile
  int kq = lane >> 4;     // 0/1: K (and M for C/D) half-select
  int t0 = wave * 16;
  int t0u = __builtin_amdgcn_readfirstlane(t0);  // wave-uniform scalar loop bound
  int trow = t0 + m;
  float wtP = Wp[trow], wtQ = Wq[trow];

  v8f accX[4] = {};
  v8f accY[4] = {};

  // --- Part 1: strictly-below-diagonal k-blocks (k0+3 < t0): no causal mask needed ---
#pragma unroll 2
  for (int k0 = 0; k0 < t0u; k0 += 4) {
    int s0 = k0 + 2 * kq;
    float ip0 = IvP[s0],     iq0 = IvQ[s0];
    float ip1 = IvP[s0 + 1], iq1 = IvQ[s0 + 1];
    v2f aP, aQ, aQn;
    aP.x = wtP * ip0 - wtQ * iq0;  aQ.x = wtP * iq0 + wtQ * ip0;
    aP.y = wtP * ip1 - wtQ * iq1;  aQ.y = wtP * iq1 + wtQ * ip1;
    aQn.x = -aQ.x; aQn.y = -aQ.y;   // f32 WMMA has no A/B neg bits
#pragma unroll
    for (int j = 0; j < 4; ++j) {
      int d0 = j * 16;
      v2f bx, by;
      bx.x = Ux[s0][d0 + m];  bx.y = Ux[s0 + 1][d0 + m];
      by.x = Uy[s0][d0 + m];  by.y = Uy[s0 + 1][d0 + m];
      cwmma(aP, aQ, aQn, bx, by, accX[j], accY[j]);
    }
  }

  // --- Part 2: diagonal 16x16 block (k0 in [t0, t0+16)): causal mask, fully unrolled ---
#pragma unroll
  for (int kk = 0; kk < 16; kk += 4) {
    int k0 = t0u + kk;
    int s0 = k0 + 2 * kq;
    float ip0 = IvP[s0],     iq0 = IvQ[s0];
    float ip1 = IvP[s0 + 1], iq1 = IvQ[s0 + 1];
    v2f aP, aQ, aQn;
    aP.x = wtP * ip0 - wtQ * iq0;  aQ.x = wtP * iq0 + wtQ * ip0;
    aP.y = wtP * ip1 - wtQ * iq1;  aQ.y = wtP * iq1 + wtQ * ip1;
    if (s0 > trow)     { aP.x = 0.0f; aQ.x = 0.0f; }
    if (s0 + 1 > trow) { aP.y = 0.0f; aQ.y = 0.0f; }
    aQn.x = -aQ.x; aQn.y = -aQ.y;
#pragma unroll
    for (int j = 0; j < 4; ++j) {
      int d0 = j * 16;
      v2f bx, by;
      bx.x = Ux[s0][d0 + m];  bx.y = Ux[s0 + 1][d0 + m];
      by.x = Uy[s0][d0 + m];  by.y = Uy[s0 + 1][d0 + m];
      cwmma(aP, aQ, aQn, bx, by, accX[j], accY[j]);
    }
  }

  // Epilogue: Y[t,d] = local_h + W_t o chunk_state ; store (re,im) pairs
#pragma unroll
  for (int j = 0; j < 4; ++j) {
    int d = j * 16 + m;
    float sx = Sx[d], sy = Sy[d];
#pragma unroll
    for (int i = 0; i < 8; ++i) {
      int t = t0 + i + 8 * kq;                 // C/D layout: VGPR i -> M = i (+8 for lanes 16..31)
      float P = Wp[t], Q = Wq[t];
      float yx = accX[j][i] + ( P * sx + Q * sy);
      float yy = accY[j][i] + (-Q * sx + P * sy);
      int base = (((b * L_ + t0g + t) * H_ + h) * D_ + d) * 2;
      v2f out; out.x = yx; out.y = yy;
      *(v2f*)(Y + base) = out;
    }
  }
}

extern "C" void kernel_launch(void* const* d_in, const int* in_sizes, int n_in,
                              void* d_out, int out_size, void* d_ws, size_t ws_size,
                              hipStream_t stream) {
  (void)in_sizes; (void)n_in; (void)out_size; (void)ws_size;
  const float* alpha = (const float*)d_in[0];
  const float* omega = (const float*)d_in[1];
  const float* dt    = (const float*)d_in[2];
  const float* K     = (const float*)d_in[3];
  const float* V     = (const float*)d_in[4];
  float* Y = (float*)d_out;

  // workspace carve (floats): totw 8192 | finh 524288 | cstate 524288  (~4.2 MB)
  float* totw   = (float*)d_ws;
  float* finh   = totw + 2 * B_ * NCH_ * H_;
  float* cstate = finh + 2 * B_ * NCH_ * H_ * D_;

  ssd_phaseA<<<B_ * NCH_ * H_, C_,  0, stream>>>(alpha, omega, dt, K, V, totw, finh);
  ssd_phaseB<<<B_ * H_,        D_,  0, stream>>>(totw, finh, cstate);
  ssd_phaseC<<<B_ * NCH_ * H_, 128, 0, stream>>>(alpha, omega, dt, K, V, cstate, Y);
}